// kernel_generated_1_62904091017320
// MI455X (gfx1250) — compile-verified
//
#include <hip/hip_runtime.h>

typedef float v2f __attribute__((ext_vector_type(2)));
typedef float v8f __attribute__((ext_vector_type(8)));

#define CIN   128
#define COUT  32
#define HH    28
#define WW    28
#define NPIX  (HH * WW)          // 784 pixels per (b, channel) image
#define GPIX  64                 // pixels handled per workgroup (4 tiles of 16)
#define XTILE (GPIX + 2)         // staged pixels incl. 1-pixel halo each side
#define XP    132                // x LDS pitch (floats per pixel), bank-conflict-free
#define WP    132                // w LDS pitch (floats per (k,co) row)
#define SX_FLOATS (XTILE * XP)       // 66*132 = 8712
#define SW_FLOATS (3 * 16 * WP)      // 48*132 = 6336  (one 16-row Cout tile)

// gfx1250 async global->LDS staging if the toolchain exposes it; otherwise
// fall back to plain load + ds_store (identical to the proven baseline).
#if __has_builtin(__builtin_amdgcn_global_load_async_to_lds_b32)
#define USE_ASYNC_LDS 1
#else
#define USE_ASYNC_LDS 0
#endif

#define AS1 __attribute__((address_space(1)))
#define AS3 __attribute__((address_space(3)))

__device__ __forceinline__ void stage_b32(const float* g, float* l) {
#if USE_ASYNC_LDS
    // GLOBAL_LOAD_ASYNC_TO_LDS_B32: copies 4 bytes from the global address g to the LDS address l; tracked by ASYNCcnt.
    // Builtin prototype (per hipcc diagnostic): (int AS1*, int AS3*, imm, imm).
    __builtin_amdgcn_global_load_async_to_lds_b32(
        (AS1 int*)g, (AS3 int*)l, /*offset=*/0, /*cpol=*/0);
#else
    *l = *g;
#endif
}

__device__ __forceinline__ void stage_wait() {
#if USE_ASYNC_LDS
#if __has_builtin(__builtin_amdgcn_s_wait_asynccnt)
    __builtin_amdgcn_s_wait_asynccnt(0);
#else
    asm volatile("s_wait_asynccnt 0" ::: "memory");
#endif
#endif
}

// D = A(16x4 f32) x B(4x16 f32) + C(16x16 f32), wave32 WMMA.
// Each wave computes one 16(Cout) x 16(pixel) output tile, reducing over
// K = Cin(128) x taps(3) = 384 in chunks of 4.
__global__ __launch_bounds__(128)
void conv1d_roll_wmma_f32(const float* __restrict__ x,
                          const float* __restrict__ w,
                          float* __restrict__ out) {
    __shared__ float sx[SX_FLOATS];  // [pix_local 0..65][ci 0..127], pitch XP
    __shared__ float sw[SW_FLOATS];  // [k 0..2][co 0..15][ci 0..127], pitch WP

    const int tid   = threadIdx.x;        // 128 threads = 4 waves
    const int grp   = blockIdx.x;         // 0..12  pixel group (64 pixels each)
    const int cb    = blockIdx.y * 16;    // Cout tile base: 0 or 16
    const int b     = blockIdx.z;         // batch
    const int pbase = grp * GPIX;

    // ---- stage x tile: pixels [pbase-1, pbase+GPIX], all 128 ci (zero OOB) ----
    const float* xb = x + (size_t)b * CIN * NPIX;
    for (int idx = tid; idx < XTILE * CIN; idx += 128) {
        const int pixl = idx % XTILE;          // consecutive tids -> consecutive
        const int ci   = idx / XTILE;          // global pixels (coalesced)
        const int pg   = pbase - 1 + pixl;
        float* dst = &sx[pixl * XP + ci];
        if (pg >= 0 && pg < NPIX) {
            stage_b32(xb + (size_t)ci * NPIX + pg, dst);
        } else {
            *dst = 0.0f;                       // halo / overhang zero-fill
        }
    }
    // ---- stage weights for this Cout tile, re-laid out as [k][co][ci] ----
    for (int idx = tid; idx < 16 * CIN * 3; idx += 128) {
        const int k  = idx % 3;                // global w is (co,ci,k): coalesced read
        const int ci = (idx / 3) % CIN;
        const int co = idx / (3 * CIN);
        stage_b32(w + ((size_t)(cb + co) * CIN + ci) * 3 + k,
                  &sw[(k * 16 + co) * WP + ci]);
    }
    stage_wait();          // drain ASYNCcnt before making LDS visible
    __syncthreads();       // also orders the plain zero-fill ds_stores

    // ---- per-wave 16x16 tile ----
    const int wv   = tid >> 5;           // 0..3: pixel tile within group
    const int lane = tid & 31;
    const int n    = lane & 15;          // N index (pixel) / A row (co)
    const int half = lane >> 4;          // K half: lanes 16-31 hold K = +2,+3
    const int kofs = 2 * half;

    const int pl   = wv * 16 + n;        // pixel within group (0..63)
    const int pg   = pbase + pl;         // global flattened pixel
    const int wcol = pg % WW;

    v8f acc = {};
    const v2f vzero = {0.0f, 0.0f};

#pragma unroll
    for (int k = 0; k < 3; ++k) {
        // tap k reads pixel (pg + k - 1); invalid across W row boundaries.
        const bool valid = (k == 1) ? true : ((k == 0) ? (wcol != 0) : (wcol != WW - 1));
        // LDS pixel slot for tap k: 1 + pl + (k-1) = pl + k  (halo at slot 0 / 65)
        const float* bx = sx + (pl + k) * XP + kofs;
        const float* aw = sw + (k * 16 + n) * WP + kofs;
#pragma unroll 8
        for (int c = 0; c < CIN; c += 4) {
            v2f afrag = *(const v2f*)(aw + c);   // A: (M=n, K=kofs+{0,1})
            v2f bfrag = *(const v2f*)(bx + c);   // B: (K=kofs+{0,1}, N=n)
            bfrag = valid ? bfrag : vzero;       // branch-free edge zeroing
            acc = __builtin_amdgcn_wmma_f32_16x16x4_f32(
                /*neg_a=*/false, afrag, /*neg_b=*/false, bfrag,
                /*c_mod=*/(short)0, acc, /*reuse_a=*/false, /*reuse_b=*/false);
        }
    }

    // ---- store with H-roll(+1): out[b, co, (h+1)%28, w] = y[b, co, h, w] ----
    if (pg < NPIX) {
        const int hrow = pg / WW;
        int hout = hrow + 1;
        if (hout == HH) hout = 0;
        const int pout = hout * WW + wcol;
        float* ob = out + ((size_t)b * COUT + cb) * NPIX + pout;
#pragma unroll
        for (int r = 0; r < 8; ++r) {
            const int M = r + 8 * half;          // C/D layout: VGPR r, halves M / M+8
            ob[(size_t)M * NPIX] = acc[r];
        }
    }
}

extern "C" void kernel_launch(void* const* d_in, const int* in_sizes, int n_in,
                              void* d_out, int out_size, void* d_ws, size_t ws_size,
                              hipStream_t stream) {
    const float* x = (const float*)d_in[0];   // (128,128,28,28) f32
    const float* w = (const float*)d_in[1];   // (32,128,3) f32
    float* out = (float*)d_out;               // (128,32,28,28) f32

    dim3 grid((NPIX + GPIX - 1) / GPIX, COUT / 16, 128);  // (13, 2, 128)
    conv1d_roll_wmma_f32<<<grid, dim3(128), 0, stream>>>(x, w, out);
}